// DynamicInstanceSegmentationLoss_50268297232942
// MI455X (gfx1250) — compile-verified
//
#include <hip/hip_runtime.h>

typedef float v2f __attribute__((ext_vector_type(2)));
typedef float v8f __attribute__((ext_vector_type(8)));

#define N_ELEM        4096
#define N_TILES       256      // 4096 / 16
#define WAVES_PER_BLK 8
#define BLK_THREADS   256

// Reference constants: A=-0.565, B=4.0, sqrt(pi)=1.772453851
// dirac(d) = c*exp(-(k*d)^2) + 1,  c = 1/(A*sqrt(pi)), k = B/A.  The +1 cancels.
// exp(-k^2 * d2) == exp2(C_EXP * d2) with C_EXP = -k^2 * log2(e)
constexpr double kA   = -0.565;
constexpr double kB   = 4.0;
constexpr double kSP  = 1.772453851;
constexpr double kK   = kB / kA;
constexpr float  C_EXP = (float)(-(kK * kK) * 1.4426950408889634);
constexpr double kC   = 1.0 / (kA * kSP);
constexpr float  SCALE = (float)((kC * kC) / (4.0 * 4096.0 * 4096.0)); // c^2 / (B*N*N)

__global__ __launch_bounds__(BLK_THREADS)
void dirac_loss_tiles(const float* __restrict__ x,
                      const float* __restrict__ t,
                      float* __restrict__ partial)
{
    __shared__ float sx[N_ELEM];
    __shared__ float st[N_ELEM];
    __shared__ float wsum[WAVES_PER_BLK];

    const int tid   = threadIdx.x;
    const int batch = blockIdx.x >> 8;    // 4 batches
    const int trow  = blockIdx.x & 255;   // 256 tile rows per batch
    const int base  = batch * N_ELEM;

    // Stage this batch's rows (16KB x 2) into LDS once
    for (int i = tid; i < N_ELEM; i += BLK_THREADS) {
        sx[i] = x[base + i];
        st[i] = t[base + i];
    }
    __syncthreads();

    const int  wave = tid >> 5;
    const int  lane = tid & 31;
    const int  l15  = lane & 15;
    const bool hiHalf = (lane >= 16);

    // A matrix (16x4, f32): row i = [x_i, x_i^2, 1, 0]
    // Lanes 0-15 : VGPR0 = K0 (x_i),  VGPR1 = K1 (x_i^2)
    // Lanes 16-31: VGPR0 = K2 (1.0),  VGPR1 = K3 (0.0)
    const int i0 = trow * 16;
    const float xi = sx[i0 + l15];
    const float ti = st[i0 + l15];
    v2f a_p, a_t;
    a_p.x = hiHalf ? 1.0f : xi;
    a_p.y = hiHalf ? 0.0f : xi * xi;
    a_t.x = hiHalf ? 1.0f : ti;
    a_t.y = hiHalf ? 0.0f : ti * ti;

    float acc = 0.0f;

    for (int jt = wave; jt < N_TILES; jt += WAVES_PER_BLK) {
        const int j0 = jt * 16;
        const float xj = sx[j0 + l15];
        const float tj = st[j0 + l15];

        // B matrix (4x16, f32): col j = [-2*x_j, 1, x_j^2, 0]^T
        // Lanes 0-15 : VGPR0 = row K0 (-2*x_j), VGPR1 = row K1 (1.0)
        // Lanes 16-31: VGPR0 = row K2 (x_j^2),  VGPR1 = row K3 (0.0)
        v2f b_p, b_t;
        b_p.x = hiHalf ? xj * xj : -2.0f * xj;
        b_p.y = hiHalf ? 0.0f    : 1.0f;
        b_t.x = hiHalf ? tj * tj : -2.0f * tj;
        b_t.y = hiHalf ? 0.0f    : 1.0f;

        v8f c0 = {};
        // D_ij = x_i^2 - 2*x_i*x_j + x_j^2 = (x_i - x_j)^2 for a full 16x16 tile
        v8f d2p = __builtin_amdgcn_wmma_f32_16x16x4_f32(
            false, a_p, false, b_p, (short)0, c0, false, false);
        v8f d2t = __builtin_amdgcn_wmma_f32_16x16x4_f32(
            false, a_t, false, b_t, (short)0, c0, false, false);

#pragma unroll
        for (int k = 0; k < 8; ++k) {
            const float ep = __builtin_amdgcn_exp2f(C_EXP * d2p[k]);
            const float et = __builtin_amdgcn_exp2f(C_EXP * d2t[k]);
            const float df = ep - et;
            acc += df * df;
        }
    }

    // Deterministic wave32 tree reduction
#pragma unroll
    for (int off = 16; off > 0; off >>= 1)
        acc += __shfl_down(acc, off, 32);
    if (lane == 0) wsum[wave] = acc;
    __syncthreads();

    if (tid == 0) {
        float s = 0.0f;
#pragma unroll
        for (int w = 0; w < WAVES_PER_BLK; ++w) s += wsum[w];
        partial[blockIdx.x] = s;
    }
}

__global__ __launch_bounds__(BLK_THREADS)
void reduce_partials(const float* __restrict__ partial,
                     float* __restrict__ out, int n)
{
    __shared__ float sm[BLK_THREADS];
    const int tid = threadIdx.x;
    float s = 0.0f;
    for (int i = tid; i < n; i += BLK_THREADS) s += partial[i];
    sm[tid] = s;
    __syncthreads();
#pragma unroll
    for (int stride = BLK_THREADS / 2; stride > 0; stride >>= 1) {
        if (tid < stride) sm[tid] += sm[tid + stride];
        __syncthreads();
    }
    if (tid == 0) out[0] = sm[0] * SCALE;
}

extern "C" void kernel_launch(void* const* d_in, const int* in_sizes, int n_in,
                              void* d_out, int out_size, void* d_ws, size_t ws_size,
                              hipStream_t stream)
{
    const float* x = (const float*)d_in[0];   // input  [4, 4096] f32
    const float* t = (const float*)d_in[1];   // target [4, 4096] f32
    float* out     = (float*)d_out;           // scalar f32
    float* partial = (float*)d_ws;            // 1024 f32 partials

    const int nblocks = 4 * N_TILES;          // 1024: one block per (batch, tile-row)
    dirac_loss_tiles<<<nblocks, BLK_THREADS, 0, stream>>>(x, t, partial);
    reduce_partials<<<1, BLK_THREADS, 0, stream>>>(partial, out, nblocks);
}